// S2Convolution_80848464380345
// MI455X (gfx1250) — compile-verified
//
#include <hip/hip_runtime.h>
#include <math.h>

typedef float v2f __attribute__((ext_vector_type(2)));
typedef float v8f __attribute__((ext_vector_type(8)));

#define PI_D 3.14159265358979323846

// ---- static configuration (matches reference) ----
#define BATCH   8
#define F_IN    64
#define F_OUT   128
#define B_IN    32
#define B_OUT   16
#define NL      16
#define MDIM    31
#define NGRID   32
#define NO_DIM  3968          // MDIM * F_OUT
#define SCALING 0.0027621358640644861   // 1/sqrt(32*64*16^4/32^2) = 1/sqrt(2^17)

// ---- workspace layout (float element offsets) ----
#define OFF_WANA 0ull
#define SZ_WANA  31744ull            // [64 j][16 l][31 m]
#define OFF_DSYN (OFF_WANA + SZ_WANA)
#define SZ_DSYN  492032ull           // [16 l][32 k][31 m][31 n]
#define OFF_FKCR (OFF_DSYN + SZ_DSYN)
#define SZ_FKC   15872ull            // [32 g][16 l][31 n]
#define OFF_FKCI (OFF_FKCR + SZ_FKC)
#define OFF_C64  (OFF_FKCI + SZ_FKC)
#define SZ_C64   1984ull             // [31 m][64 a]
#define OFF_S64  (OFF_C64 + SZ_C64)
#define OFF_CC   (OFF_S64 + SZ_C64)  // [32 n][32 c]
#define OFF_SC   (OFF_CC  + 1024ull)
#define OFF_CAT  (OFF_SC  + 1024ull) // [32 a][32 m]
#define OFF_SAT  (OFF_CAT + 1024ull)
#define OFF_XM   (OFF_SAT + 1024ull)
#define SZ_XM    2031616ull          // [31 m][512 bf][64 j][2]
#define OFF_XHR  (OFF_XM + SZ_XM)
#define SZ_XH    262144ull           // [16 l][256 mb][64 i]
#define OFF_XHI  (OFF_XHR + SZ_XH)
#define OFF_KHR  (OFF_XHI + SZ_XH)
#define SZ_KH    4063232ull          // [16 l][64 i][3968 no]
#define OFF_KHI  (OFF_KHR + SZ_KH)
#define OFF_Z2   (OFF_KHI + SZ_KH)
#define SZ_Z2    31490048ull         // [1024 bo][16 l][961 mn][2]

// ================= device helpers =================

__device__ __forceinline__ void build_lf(double* LF) {
  if (threadIdx.x == 0) {
    double r = 0.0;
    LF[0] = 0.0;
    for (int i = 1; i < 40; ++i) { r += log((double)i); LF[i] = r; }
  }
  __syncthreads();
}

// Wigner small-d element d^l_{m1,m2}(beta), double precision
__device__ double wig_d(const double* LF, int l, int m1, int m2, double beta) {
  double c = cos(0.5 * beta), s = sin(0.5 * beta);
  double lc = log(c), ls = log(s);
  double pre = 0.5 * (LF[l + m1] + LF[l - m1] + LF[l + m2] + LF[l - m2]);
  double sum = 0.0;
  for (int k = 0; k <= 2 * l; ++k) {
    int a1 = l + m2 - k, a3 = m1 - m2 + k, a4 = l - m1 - k;
    if ((a1 | a3 | a4) < 0) continue;
    double t = exp(pre - LF[a1] - LF[k] - LF[a3] - LF[a4]
                   + (double)(a1 + a4) * lc + (double)(a3 + k) * ls);
    sum += (a3 & 1) ? -t : t;
  }
  return sum;
}

// f32 WMMA 16x16x4 wrapper (D = A*B + C)
__device__ __forceinline__ v8f wmma4(v2f a, v2f b, v8f c) {
  return __builtin_amdgcn_wmma_f32_16x16x4_f32(false, a, false, b, (short)0, c,
                                               false, false);
}
// A fragment: lane L<16 -> row rbase+L, cols k0..k0+1; lane>=16 -> cols k0+2..3
__device__ __forceinline__ v2f ldA(const float* p, int ld, int rbase, int k0,
                                   int lrow, int kh) {
  const float* q = p + (size_t)(rbase + lrow) * ld + (k0 + 2 * kh);
  v2f r; r.x = q[0]; r.y = q[1]; return r;
}
// B fragment: rows k0+2*kh, k0+2*kh+1 ; col cbase+lrow
__device__ __forceinline__ v2f ldB(const float* p, int ld, int k0, int cbase,
                                   int lrow, int kh) {
  const float* q = p + (size_t)(k0 + 2 * kh) * ld + (cbase + lrow);
  v2f r; r.x = q[0]; r.y = q[ld]; return r;
}

// ================= table kernels =================

__global__ void k_tab_wana(float* __restrict__ wana) {
  __shared__ double LF[40]; build_lf(LF);
  int t = blockIdx.x * blockDim.x + threadIdx.x;
  if (t >= 64 * 16 * 31) return;
  int ms = t % 31, l = (t / 31) % 16, j = t / (31 * 16);
  int mp = ms - 15;
  float val = 0.f;
  if (mp >= -l && mp <= l) {
    double b = PI_D * (2.0 * j + 0.5) / 128.0;     // betas_in
    double wsum = 0.0;
    for (int kk = 0; kk < 32; ++kk)
      wsum += sin(b * (2.0 * kk + 1.0)) / (2.0 * kk + 1.0);
    double wj = (2.0 * PI_D / 64.0) * (2.0 / 32.0) * sin(b) * wsum;
    val = (float)(wj * wig_d(LF, l, mp, 0, b));
  }
  wana[t] = val;
}

__global__ void k_tab_dsyn(float* __restrict__ dsyn) {
  __shared__ double LF[40]; build_lf(LF);
  int t = blockIdx.x * blockDim.x + threadIdx.x;
  if (t >= 16 * 32 * 31 * 31) return;
  int ns = t % 31, ms = (t / 31) % 31, kk = (t / 961) % 32, l = t / (961 * 32);
  int mp = ms - 15, np = ns - 15;
  float val = 0.f;
  if (mp >= -l && mp <= l && np >= -l && np <= l) {
    double b = PI_D * (2.0 * kk + 0.5) / 64.0;     // betas_out
    val = (float)((2.0 * l + 1.0) * wig_d(LF, l, mp, np, b));
  }
  dsyn[t] = val;
}

__global__ void k_tab_fkc(float* __restrict__ fkcr, float* __restrict__ fkci) {
  __shared__ double LF[40]; build_lf(LF);
  int t = blockIdx.x * blockDim.x + threadIdx.x;
  if (t >= 32 * 16 * 31) return;
  int ms = t % 31, l = (t / 31) % 16, g = t / (31 * 16);
  int gb = g >> 3, ga = g & 7;            // beta-major grid
  int mp = ms - 15;
  float vr = 0.f, vi = 0.f;
  if (mp >= -l && mp <= l) {
    double beta  = (PI_D / 8.0) * (double)(gb + 1) / 4.0;
    double alpha = 2.0 * PI_D * (double)ga / 8.0;
    double d = wig_d(LF, l, mp, 0, beta) * SCALING;
    vr = (float)( d * cos(mp * alpha));   // conj(FK) with SCALING folded in
    vi = (float)(-d * sin(mp * alpha));
  }
  fkcr[t] = vr; fkci[t] = vi;
}

__global__ void k_tab_trig(float* __restrict__ c64, float* __restrict__ s64,
                           float* __restrict__ cc,  float* __restrict__ sc,
                           float* __restrict__ cat_, float* __restrict__ sat_) {
  int t = blockIdx.x * blockDim.x + threadIdx.x;
  if (t < 1984) {
    int m = t / 64, a = t % 64;
    double ang = 2.0 * PI_D * (double)((m - 15) * a) / 64.0;
    c64[t] = (float)cos(ang); s64[t] = (float)sin(ang);
  } else if (t < 1984 + 1024) {
    int i = t - 1984; int n = i >> 5, c = i & 31;
    double ang = 2.0 * PI_D * (double)((n - 15) * c) / 32.0;
    cc[i] = (float)cos(ang); sc[i] = (float)sin(ang);
  } else if (t < 1984 + 2048) {
    int i = t - 1984 - 1024; int a = i >> 5, m = i & 31;
    double ang = 2.0 * PI_D * (double)((m - 15) * a) / 32.0;
    cat_[i] = (float)cos(ang); sat_[i] = (float)sin(ang);
  }
}

// ================= pipeline kernels =================

// alpha-DFT: xm[m][bf][j] = sum_a x[bf][j][a] * e^{-i*2pi*m'*a/64}
__global__ void k_dft(const float* __restrict__ x, const float* __restrict__ c64,
                      const float* __restrict__ s64, float* __restrict__ xm) {
  int t = blockIdx.x * blockDim.x + threadIdx.x;
  if (t >= 31 * 512 * 64) return;
  int j = t & 63, bf = (t >> 6) & 511, m = t >> 15;
  const float* xp = x + ((size_t)bf * 64 + j) * 64;
  const float* cp = c64 + m * 64;
  const float* sp = s64 + m * 64;
  float cr = 0.f, ci = 0.f;
  for (int a = 0; a < 64; ++a) {
    float xv = xp[a];
    cr += xv * cp[a];
    ci -= xv * sp[a];
  }
  size_t o = (size_t)t * 2;
  xm[o] = cr; xm[o + 1] = ci;
}

// S2 analysis: xhat[l][(m,b)][f] = sum_j W_ANA[j,l,m] * xm[m][bf][j]
__global__ void k_analysis(const float* __restrict__ wana,
                           const float* __restrict__ xm,
                           float* __restrict__ xhr, float* __restrict__ xhi) {
  int t = blockIdx.x * blockDim.x + threadIdx.x;
  if (t >= 16 * 256 * 64) return;
  int f = t & 63, mb = (t >> 6) & 255, l = t >> 14;
  int m = mb >> 3, b = mb & 7;
  float ar = 0.f, ai = 0.f;
  if (m < 31) {
    const float* xp = xm + (((size_t)m * 512 + (b * 64 + f)) * 64) * 2;
    for (int j = 0; j < 64; ++j) {
      float w = wana[(j * 16 + l) * 31 + m];
      ar += w * xp[2 * j];
      ai += w * xp[2 * j + 1];
    }
  }
  xhr[t] = ar; xhi[t] = ai;   // padded rows (m==31) -> zero
}

// kernel s2_rft: khc[l][i][(n,o)] = sum_g kernel[i,o,g]*conj(FK)[g,l,n]
__global__ void k_kexp(const float* __restrict__ ker,
                       const float* __restrict__ fkcr,
                       const float* __restrict__ fkci,
                       float* __restrict__ khr, float* __restrict__ khi) {
  int blk = blockIdx.x;                    // (l*64 + i)*31 + n
  int n = blk % 31, i = (blk / 31) % 64, l = blk / (31 * 64);
  int o = threadIdx.x;                     // 128 threads
  float ar = 0.f, ai = 0.f;
  int np = n - 15;
  if (np >= -l && np <= l) {
    const float* kp = ker + ((size_t)i * 128 + o) * 32;
    for (int g = 0; g < 32; ++g) {
      float kv = kp[g];
      int fi = (g * 16 + l) * 31 + n;
      ar += kv * fkcr[fi];
      ai += kv * fkci[fi];
    }
  }
  size_t idx = ((size_t)l * 64 + i) * 3968 + n * 128 + o;
  khr[idx] = ar; khi[idx] = ai;
}

// main complex GEMM per degree l: z[l][(m,b)][(n,o)] = xhat[l] * khc[l]
// one wave per 16x16 tile; K=64 -> 16 wmma k-steps x 4 real wmmas (complex)
__global__ void __launch_bounds__(256)
k_mainmm(const float* __restrict__ xhr, const float* __restrict__ xhi,
         const float* __restrict__ khr, const float* __restrict__ khi,
         float* __restrict__ z2) {
  int blk = blockIdx.x;                    // 16 l * 16 Mtiles * 31 Ngroups
  int l = blk / 496;
  int rem = blk % 496;
  int mt = rem / 31;
  int ngrp = rem % 31;
  int wave = threadIdx.x >> 5;
  int lane = threadIdx.x & 31;
  int lrow = lane & 15;
  int kh = (lane >> 4) & 1;
  int nt = ngrp * 8 + wave;                // 248 N tiles

  const float* arB = xhr + (size_t)l * 256 * 64;
  const float* aiB = xhi + (size_t)l * 256 * 64;
  const float* brB = khr + (size_t)l * 64 * 3968;
  const float* biB = khi + (size_t)l * 64 * 3968;

  v8f cr = {}; v8f ci = {};
  int rbase = mt * 16, cbase = nt * 16;
  for (int kk = 0; kk < 16; ++kk) {
    int k0 = kk * 4;
    v2f a_r = ldA(arB, 64, rbase, k0, lrow, kh);
    v2f a_i = ldA(aiB, 64, rbase, k0, lrow, kh);
    v2f b_r = ldB(brB, 3968, k0, cbase, lrow, kh);
    v2f b_i = ldB(biB, 3968, k0, cbase, lrow, kh);
    v2f a_in; a_in.x = -a_i.x; a_in.y = -a_i.y;
    cr = wmma4(a_r, b_r, cr);
    cr = wmma4(a_in, b_i, cr);   // Cr = ArBr - AiBi
    ci = wmma4(a_r, b_i, ci);
    ci = wmma4(a_i, b_r, ci);    // Ci = ArBi + AiBr
  }
  // scatter-store into [bo][l][m*31+n][2] (L2-resident; synth reads coalesced)
  int nocol = cbase + lrow;
  int n = nocol >> 7, o = nocol & 127;
  for (int v = 0; v < 8; ++v) {
    int mbrow = rbase + v + 8 * kh;
    int m = mbrow >> 3;
    if (m >= 31) continue;                 // drop M padding
    int b = mbrow & 7;
    size_t idx = ((((size_t)(b * 128 + o) * 16 + l) * 961) + m * 31 + n) * 2;
    z2[idx]     = cr[v];
    z2[idx + 1] = ci[v];
  }
}

// fused degree-sum + inverse SO(3) 2D transform + bias, one block per (b,o,k)
__global__ void __launch_bounds__(256)
k_synth(const float* __restrict__ dsyn, const float* __restrict__ z2,
        const float* __restrict__ cc,   const float* __restrict__ sc,
        const float* __restrict__ cat_, const float* __restrict__ sat_,
        const float* __restrict__ bias, float* __restrict__ out) {
  __shared__ float Fr[1024], Fi[1024], Tr[1024], Ti[1024];
  int bo = blockIdx.x >> 5;
  int k  = blockIdx.x & 31;
  int tid = threadIdx.x;

  for (int i = tid; i < 1024; i += 256) { Fr[i] = 0.f; Fi[i] = 0.f; }
  __syncthreads();

  // fmn[m,n] = sum_l D_SYN[l,k,m,n] * z[l,m,n,b,o]  (VALU, coalesced reads)
  for (int p = 0; p < 4; ++p) {
    int mn = p * 256 + tid;
    if (mn < 961) {
      float ar = 0.f, ai = 0.f;
      for (int l = 0; l < 16; ++l) {
        float d = dsyn[((size_t)(l * 32 + k)) * 961 + mn];
        size_t zi = (((size_t)bo * 16 + l) * 961 + mn) * 2;
        ar += d * z2[zi];
        ai += d * z2[zi + 1];
      }
      int m = mn / 31, n = mn % 31;
      Fr[m * 32 + n] = ar;
      Fi[m * 32 + n] = ai;
    }
  }
  __syncthreads();

  int wave = tid >> 5, lane = tid & 31;
  int lrow = lane & 15, kh = (lane >> 4) & 1;

  // stage 1: T = F * E_c  (Tr = Fr*Cc - Fi*Sc ; Ti = Fr*Sc + Fi*Cc)
  {
    int tIm = wave >> 2;                 // 0 -> Tr, 1 -> Ti
    int tm = (wave >> 1) & 1, tn = wave & 1;
    v8f acc = {};
    for (int kk = 0; kk < 8; ++kk) {
      int k0 = kk * 4;
      v2f afr = ldA(Fr, 32, tm * 16, k0, lrow, kh);
      v2f afi = ldA(Fi, 32, tm * 16, k0, lrow, kh);
      v2f bc  = ldB(cc, 32, k0, tn * 16, lrow, kh);
      v2f bs  = ldB(sc, 32, k0, tn * 16, lrow, kh);
      if (tIm == 0) {
        v2f afin; afin.x = -afi.x; afin.y = -afi.y;
        acc = wmma4(afr, bc, acc);
        acc = wmma4(afin, bs, acc);
      } else {
        acc = wmma4(afr, bs, acc);
        acc = wmma4(afi, bc, acc);
      }
    }
    float* Td = tIm ? Ti : Tr;
    for (int v = 0; v < 8; ++v)
      Td[(tm * 16 + v + 8 * kh) * 32 + tn * 16 + lrow] = acc[v];
  }
  __syncthreads();

  // stage 2: out = CaT*Tr - SaT*Ti  (+ bias), write [b,o,k,a,c]
  if (wave < 4) {
    int tm = (wave >> 1) & 1, tn = wave & 1;
    v8f acc = {};
    for (int kk = 0; kk < 8; ++kk) {
      int k0 = kk * 4;
      v2f aC = ldA(cat_, 32, tm * 16, k0, lrow, kh);
      v2f aS = ldA(sat_, 32, tm * 16, k0, lrow, kh);
      v2f aSn; aSn.x = -aS.x; aSn.y = -aS.y;
      v2f bR = ldB(Tr, 32, k0, tn * 16, lrow, kh);
      v2f bI = ldB(Ti, 32, k0, tn * 16, lrow, kh);
      acc = wmma4(aC, bR, acc);
      acc = wmma4(aSn, bI, acc);
    }
    float bv = bias[bo & 127];
    size_t base = ((size_t)bo * 32 + k) * 1024;
    int c = tn * 16 + lrow;
    for (int v = 0; v < 8; ++v) {
      int a = tm * 16 + v + 8 * kh;
      out[base + a * 32 + c] = acc[v] + bv;
    }
  }
}

// ================= host launcher =================

extern "C" void kernel_launch(void* const* d_in, const int* in_sizes, int n_in,
                              void* d_out, int out_size, void* d_ws,
                              size_t ws_size, hipStream_t stream) {
  (void)in_sizes; (void)n_in; (void)out_size; (void)ws_size;
  const float* x    = (const float*)d_in[0];
  const float* ker  = (const float*)d_in[1];
  const float* bias = (const float*)d_in[2];
  float* out = (float*)d_out;
  float* ws  = (float*)d_ws;

  float* wana = ws + OFF_WANA;
  float* dsyn = ws + OFF_DSYN;
  float* fkcr = ws + OFF_FKCR;
  float* fkci = ws + OFF_FKCI;
  float* c64  = ws + OFF_C64;
  float* s64  = ws + OFF_S64;
  float* cc   = ws + OFF_CC;
  float* sc   = ws + OFF_SC;
  float* cat_ = ws + OFF_CAT;
  float* sat_ = ws + OFF_SAT;
  float* xm   = ws + OFF_XM;
  float* xhr  = ws + OFF_XHR;
  float* xhi  = ws + OFF_XHI;
  float* khr  = ws + OFF_KHR;
  float* khi  = ws + OFF_KHI;
  float* z2   = ws + OFF_Z2;

  hipLaunchKernelGGL(k_tab_wana, dim3(124),  dim3(256), 0, stream, wana);
  hipLaunchKernelGGL(k_tab_dsyn, dim3(1922), dim3(256), 0, stream, dsyn);
  hipLaunchKernelGGL(k_tab_fkc,  dim3(62),   dim3(256), 0, stream, fkcr, fkci);
  hipLaunchKernelGGL(k_tab_trig, dim3(16),   dim3(256), 0, stream,
                     c64, s64, cc, sc, cat_, sat_);

  hipLaunchKernelGGL(k_dft,      dim3(3968), dim3(256), 0, stream,
                     x, c64, s64, xm);
  hipLaunchKernelGGL(k_analysis, dim3(1024), dim3(256), 0, stream,
                     wana, xm, xhr, xhi);
  hipLaunchKernelGGL(k_kexp,     dim3(31744), dim3(128), 0, stream,
                     ker, fkcr, fkci, khr, khi);
  hipLaunchKernelGGL(k_mainmm,   dim3(7936), dim3(256), 0, stream,
                     xhr, xhi, khr, khi, z2);
  hipLaunchKernelGGL(k_synth,    dim3(32768), dim3(256), 0, stream,
                     dsyn, z2, cc, sc, cat_, sat_, bias, out);
}